// MixtureOfMemory_24215025615155
// MI455X (gfx1250) — compile-verified
//
#include <hip/hip_runtime.h>

// MOM step on MI455X (gfx1250, wave32).
// Bandwidth-bound: 128 MiB state read + 128 MiB state write => ~11.5us floor
// at 23.3 TB/s. Streaming update uses B128 global loads/stores; the q^T * M
// readout is computed with V_WMMA_F32_16X16X4_F32 on LDS-staged panels.

#define N_MEM 8
#define BATCH 64
#define DK 256
#define DV 256
#define LDS_STRIDE 272  // padded row stride (floats): rows k and k+2 hit disjoint banks

typedef __attribute__((ext_vector_type(2))) float v2f;
typedef __attribute__((ext_vector_type(8))) float v8f;

__global__ __launch_bounds__(256) void mom_step_kernel(
    const float* __restrict__ states, const float* __restrict__ key,
    const float* __restrict__ value, const float* __restrict__ alpha,
    const float* __restrict__ rho, const float* __restrict__ lam,
    const float* __restrict__ query, float* __restrict__ out_states,
    float* __restrict__ out_read) {
  __shared__ __align__(16) float s_panel[32 * LDS_STRIDE];  // 34 KiB newS panel
  __shared__ float s_q[DK];
  __shared__ float s_key[DK];
  __shared__ float s_val[DV];

  const int tid = threadIdx.x;
  const int bid = blockIdx.x;  // bid = n*B + b (matches states flat layout)
  const int b = bid & (BATCH - 1);
  const int n = bid >> 6;

  // Stage per-b vectors into LDS (one element per thread; DK==DV==256).
  s_q[tid]   = query[b * DK + tid];
  s_key[tid] = key[b * DK + tid];
  s_val[tid] = value[b * DV + tid];

  const float lam_s = lam[b * N_MEM + n];
  const float w_s   = rho[b * N_MEM + n] * alpha[b];

  const float* Sg = states + (size_t)bid * (DK * DV);
  float* Og = out_states + (size_t)bid * (DK * DV);

  __syncthreads();

  const int lane = tid & 31;
  const int wave = tid >> 5;        // 8 waves; wave handles v-tiles 2w, 2w+1
  const int half = lane >> 4;       // 0: lanes 0-15, 1: lanes 16-31
  const int ln   = lane & 15;
  const int v0a  = (2 * wave) * 16;
  const int v0b  = (2 * wave + 1) * 16;

  v8f c0 = {};  // readout accumulator, v-tile 2w
  v8f c1 = {};  // readout accumulator, v-tile 2w+1

  for (int p = 0; p < 8; ++p) {  // 8 panels of 32 rows (k)
    // ---- Stream panel: newS = lam*S + (rho*alpha)*key[k]*value[v] ----
    // Panel = 32 rows x 64 float4 = 2048 float4; 8 per thread, coalesced.
#pragma unroll
    for (int it = 0; it < 8; ++it) {
      const int idx = it * 256 + tid;  // 0..2047
      const int row = idx >> 6;        // 0..31
      const int v = (idx & 63) * 4;
      const int goff = (p * 32 + row) * DV + v;
      float4 s4 = *(const float4*)(Sg + goff);
      float4 vv = *(const float4*)(s_val + v);
      const float kw = s_key[p * 32 + row] * w_s;
      float4 ns;
      ns.x = lam_s * s4.x + kw * vv.x;
      ns.y = lam_s * s4.y + kw * vv.y;
      ns.z = lam_s * s4.z + kw * vv.z;
      ns.w = lam_s * s4.w + kw * vv.w;
      *(float4*)(Og + goff) = ns;                         // global_store_b128
      *(float4*)(s_panel + row * LDS_STRIDE + v) = ns;    // ds_store_b128
    }
    __syncthreads();

    // ---- WMMA readout: accumulate q[k] * newS[k,v] over this panel ----
    // A[m,k] = q[kb+k] replicated over m (all 16 D rows identical).
    // A f32 16x4 layout: VGPR0/1 = K{0,1} for lanes 0-15, K{2,3} for 16-31.
    // B f32 4x16 layout: lane.N = lane&15; VGPR0/1 = K{0,1}/{2,3} by half.
#pragma unroll
    for (int kk = 0; kk < 8; ++kk) {  // 8 k-steps of K=4 cover 32 rows
      const int kb = kk * 4 + half * 2;
      v2f a;
      a.x = s_q[p * 32 + kb + 0];
      a.y = s_q[p * 32 + kb + 1];
      const int r0 = kb * LDS_STRIDE;
      v2f b0, b1;
      b0.x = s_panel[r0 + v0a + ln];
      b0.y = s_panel[r0 + LDS_STRIDE + v0a + ln];
      b1.x = s_panel[r0 + v0b + ln];
      b1.y = s_panel[r0 + LDS_STRIDE + v0b + ln];
      c0 = __builtin_amdgcn_wmma_f32_16x16x4_f32(false, a, false, b0,
                                                 (short)0, c0, false, false);
      c1 = __builtin_amdgcn_wmma_f32_16x16x4_f32(false, a, false, b1,
                                                 (short)0, c1, false, false);
    }
    __syncthreads();  // protect LDS panel before next overwrite
  }

  // D 16x16 f32 layout: VGPR0, lanes 0-15 hold (M=0, N=lane); every D row is
  // the same readout vector, so row 0 suffices.
  if (lane < 16) {
    float* rp = out_read + (size_t)bid * DV;
    rp[v0a + lane] = c0[0];
    rp[v0b + lane] = c1[0];
  }
}

extern "C" void kernel_launch(void* const* d_in, const int* in_sizes, int n_in,
                              void* d_out, int out_size, void* d_ws,
                              size_t ws_size, hipStream_t stream) {
  (void)in_sizes; (void)n_in; (void)out_size; (void)d_ws; (void)ws_size;
  const float* states = (const float*)d_in[0];
  const float* key    = (const float*)d_in[1];
  const float* value  = (const float*)d_in[2];
  const float* alpha  = (const float*)d_in[3];
  const float* rho    = (const float*)d_in[4];
  const float* lam    = (const float*)d_in[5];
  const float* query  = (const float*)d_in[6];

  float* out_states = (float*)d_out;
  float* out_read   = out_states + (size_t)N_MEM * BATCH * DK * DV;

  hipLaunchKernelGGL(mom_step_kernel, dim3(N_MEM * BATCH), dim3(256), 0,
                     stream, states, key, value, alpha, rho, lam, query,
                     out_states, out_read);
}